// EdgeConv_54417235640997
// MI455X (gfx1250) — compile-verified
//
#include <hip/hip_runtime.h>
#include <hip/hip_bf16.h>

// EdgeConv (k=10 exact KNN + shared linear + neighbor max) for MI455X gfx1250.
//
// Decomposition:
//   sq[n]   = |pts[n]|^2                                   (tiny kernel)
//   Z[n][o] = dot(pts[n], W[o]) + b[o]                     (tiny GEMM kernel)
//   Gram    = pts @ pts^T via V_WMMA_F32_16X16X4_F32, fused with per-row
//             top-10-min on key = sq[col] - 2*dot (row-constant sq dropped),
//             then out[n][o] = max_k Z[idx[n][k]][o].
//
// Each wave32 owns 32 rows (two 16-row A tiles) so every B operand feeds TWO
// WMMA chains: 1 load per WMMA, half the L2 traffic, 2x WMMA ILP per wave.
//
// x memory layout is (1, D, N): element (d, n) at x[d*N + n] -> pts[n][d].

#define NPTS 16384
#define DIM 64
#define OCH 128
#define KNN 10

typedef __attribute__((ext_vector_type(2))) float v2f;
typedef __attribute__((ext_vector_type(8))) float v8f;

// ---------------- sq[n] = sum_d pts[n][d]^2 ----------------
__global__ __launch_bounds__(256) void sq_kernel(const float* __restrict__ x,
                                                 float* __restrict__ sq) {
  int n = blockIdx.x * 256 + threadIdx.x;
  float s = 0.0f;
  #pragma unroll 8
  for (int d = 0; d < DIM; ++d) {
    float v = x[d * NPTS + n];   // coalesced: n contiguous across lanes
    s = fmaf(v, v, s);
  }
  sq[n] = s;
}

// ---------------- Z[n][o] = dot(pts[n], W[o]) + b[o] ----------------
__global__ __launch_bounds__(128) void z_kernel(const float* __restrict__ x,
                                                const float* __restrict__ W,
                                                const float* __restrict__ b,
                                                float* __restrict__ Z) {
  int o = threadIdx.x;   // 0..127
  int n = blockIdx.x;    // 0..16383
  float acc = b[o];
  #pragma unroll 8
  for (int d = 0; d < DIM; ++d) {
    // x[d*N+n] is wave-uniform -> scalar load; W stays hot in L0/L2 (32 KB)
    acc = fmaf(x[d * NPTS + n], W[o * DIM + d], acc);
  }
  Z[n * OCH + o] = acc;  // coalesced over o
}

// Insert candidate into a 10-slot unsorted list (LDS); returns new threshold
// (= max of list). Only called when key beats the current register threshold.
__device__ __forceinline__ float insert_cand(float* cdp, unsigned short* cip,
                                             float key, int col) {
  int worst = 0;
  float wv = cdp[0];
  #pragma unroll
  for (int k = 1; k < KNN; ++k) {
    float v = cdp[k];
    if (v > wv) { wv = v; worst = k; }
  }
  cdp[worst] = key;
  cip[worst] = (unsigned short)col;
  float nv = cdp[0];
  #pragma unroll
  for (int k = 1; k < KNN; ++k) {
    float v = cdp[k];
    nv = v > nv ? v : nv;
  }
  return nv;
}

// ---------------- fused Gram-GEMM + top-10 + gather-max ----------------
__global__ __launch_bounds__(32) void knn_edgeconv_kernel(
    const float* __restrict__ x, const float* __restrict__ sq,
    const float* __restrict__ Z, float* __restrict__ out) {
  __shared__ float cd[32][16][KNN];           // 20 KB: per-(row,lane) top-10 keys
  __shared__ unsigned short ci[32][16][KNN];  // 10 KB: matching column indices
  __shared__ unsigned short sel[32][KNN];     // final 10 neighbors per row

  const int lane = threadIdx.x;  // 0..31 (wave32)
  const int lo = lane & 15;      // column-within-tile (B/C/D layouts)
  const int hi = lane >> 4;      // half-select (C/D: M+8; A/B: K+2)
  const int m0 = blockIdx.x * 32;

  const float FINF = 3.0e38f;

  // init candidate lists (single wave per WG; LDS is in-order per wave)
  #pragma unroll
  for (int r = 0; r < 8; ++r) {
    int lr = r + 8 * hi;
    #pragma unroll
    for (int k = 0; k < KNN; ++k) {
      cd[lr][lo][k] = FINF;       ci[lr][lo][k] = 0;
      cd[lr + 16][lo][k] = FINF;  ci[lr + 16][lo][k] = 0;
    }
  }
  float thr0[8], thr1[8];
  #pragma unroll
  for (int r = 0; r < 8; ++r) { thr0[r] = FINF; thr1[r] = FINF; }

  // Two A tiles: rows [m0, m0+16) and [m0+16, m0+32), all 16 K-steps of 4.
  // ISA A 16x4 layout: VGPR0 = {K=0 | K=2}, VGPR1 = {K=1 | K=3}, M = lane%16.
  v2f A0[16], A1[16];
  #pragma unroll
  for (int kk = 0; kk < 16; ++kk) {
    int d0 = 4 * kk + 2 * hi;
    A0[kk].x = x[d0 * NPTS + m0 + lo];
    A0[kk].y = x[(d0 + 1) * NPTS + m0 + lo];
    A1[kk].x = x[d0 * NPTS + m0 + 16 + lo];
    A1[kk].y = x[(d0 + 1) * NPTS + m0 + 16 + lo];
  }

  for (int j = 0; j < NPTS / 16; ++j) {
    const int c0 = j * 16;
    v8f acc0 = {0.f, 0.f, 0.f, 0.f, 0.f, 0.f, 0.f, 0.f};
    v8f acc1 = {0.f, 0.f, 0.f, 0.f, 0.f, 0.f, 0.f, 0.f};
    #pragma unroll
    for (int kk = 0; kk < 16; ++kk) {
      int d0 = 4 * kk + 2 * hi;
      v2f B;                               // B 4x16: N = lane%16, K split by hi
      B.x = x[d0 * NPTS + c0 + lo];        // coalesced: 64B per half-wave
      B.y = x[(d0 + 1) * NPTS + c0 + lo];
      acc0 = __builtin_amdgcn_wmma_f32_16x16x4_f32(false, A0[kk], false, B,
                                                   (short)0, acc0, false, false);
      acc1 = __builtin_amdgcn_wmma_f32_16x16x4_f32(false, A1[kk], false, B,
                                                   (short)0, acc1, false, false);
    }

    // epilogue: key = sq[col] - 2*dot (row-constant term dropped; per-row
    // ranking unchanged; self-match key = -sq[n] is the guaranteed minimum)
    const float sqc = sq[c0 + lo];
    const int col = c0 + lo;
    #pragma unroll
    for (int r = 0; r < 8; ++r) {
      float key = fmaf(-2.0f, acc0[r], sqc);
      if (key < thr0[r]) {                 // rare after warm-up
        int lr = r + 8 * hi;               // C/D layout: VGPR r holds M = r+8*hi
        thr0[r] = insert_cand(&cd[lr][lo][0], &ci[lr][lo][0], key, col);
      }
    }
    #pragma unroll
    for (int r = 0; r < 8; ++r) {
      float key = fmaf(-2.0f, acc1[r], sqc);
      if (key < thr1[r]) {
        int lr = 16 + r + 8 * hi;
        thr1[r] = insert_cand(&cd[lr][lo][0], &ci[lr][lo][0], key, col);
      }
    }
  }

  // merge: per row, 16 lanes x 10 slots = 160 finite candidates -> 10 smallest
  {
    const int lr = lane;  // one lane per row, 32 rows
    for (int k = 0; k < KNN; ++k) {
      float best = FINF; int bl = 0, bs = 0;
      for (int l = 0; l < 16; ++l)
        #pragma unroll
        for (int s = 0; s < KNN; ++s) {
          float v = cd[lr][l][s];
          if (v < best) { best = v; bl = l; bs = s; }
        }
      sel[lr][k] = ci[lr][bl][bs];
      cd[lr][bl][bs] = FINF;  // consume
    }
  }

  // output: out[m0+lr][o] = max_k Z[sel[lr][k]][o]; o = lane + 32*t (coalesced)
  for (int lr = 0; lr < 32; ++lr) {
    const int row = m0 + lr;
    #pragma unroll
    for (int t = 0; t < 4; ++t) {
      const int o = lane + 32 * t;
      float m = -3.0e38f;
      #pragma unroll
      for (int k = 0; k < KNN; ++k) {
        float v = Z[(int)sel[lr][k] * OCH + o];  // sel uniform -> LDS broadcast
        m = v > m ? v : m;
      }
      out[row * OCH + o] = m;
    }
  }
}

extern "C" void kernel_launch(void* const* d_in, const int* in_sizes, int n_in,
                              void* d_out, int out_size, void* d_ws, size_t ws_size,
                              hipStream_t stream) {
  (void)in_sizes; (void)n_in; (void)out_size; (void)ws_size;
  const float* x = (const float*)d_in[0];  // (1, 64, 16384)
  const float* W = (const float*)d_in[1];  // (128, 64)
  const float* b = (const float*)d_in[2];  // (128,)
  float* out = (float*)d_out;              // (16384, 128)

  float* Z  = (float*)d_ws;                // 16384*128 floats (8 MB)
  float* sq = Z + (size_t)NPTS * OCH;      // 16384 floats (64 KB)

  sq_kernel<<<NPTS / 256, 256, 0, stream>>>(x, sq);
  z_kernel<<<NPTS, OCH, 0, stream>>>(x, W, b, Z);
  knn_edgeconv_kernel<<<NPTS / 32, 32, 0, stream>>>(x, sq, Z, out);
}